// LengthRegulator_46780783788700
// MI455X (gfx1250) — compile-verified
//
#include <hip/hip_runtime.h>
#include <stdint.h>

// LengthRegulator for MI455X (gfx1250).
// Pure data movement: ~126 MB total traffic -> ~5.4 us floor at 23.3 TB/s.
// Row payloads move via the CDNA5 async global<->LDS data mover (ASYNCcnt),
// double-buffered per wave so load and store latencies overlap; zero-fill
// rows stream from an immutable pre-zeroed LDS slot with no waits.

#define B_   32
#define T_   512
#define D_   384
#define ML_  2048
#define ROW_BYTES (D_ * 4)        // 1536 bytes per row = 3 x (32 lanes x 16B)
#define POS_PER_BLOCK 64          // 8 waves x 8 rows
#define WAVES 8

// ---- CDNA5 async data-mover primitives (inline asm; assembler-verified) ----

__device__ __forceinline__ uint32_t lds_off_u32(const void* p) {
  // generic -> LDS(AS3) addrspacecast; AS3 pointer value is the LDS byte offset
  return (uint32_t)(uintptr_t)(__attribute__((address_space(3))) const char*)p;
}

__device__ __forceinline__ void async_load_b128(uint32_t lds, uint64_t gaddr) {
  // copies 16 bytes from global address gaddr to LDS offset lds, tracked by ASYNCcnt
  asm volatile("global_load_async_to_lds_b128 %0, %1, off"
               :: "v"(lds), "v"(gaddr) : "memory");
}

__device__ __forceinline__ void async_store_b128(uint64_t gaddr, uint32_t lds) {
  // copies 16 bytes from LDS offset lds to global address gaddr, tracked by ASYNCcnt
  asm volatile("global_store_async_from_lds_b128 %0, %1, off"
               :: "v"(gaddr), "v"(lds) : "memory");
}

__device__ __forceinline__ void wait_async0() {
  asm volatile("s_wait_asynccnt 0x0" ::: "memory");
}

__device__ __forceinline__ void async_load_row(uint32_t lds, uint64_t g) {
  async_load_b128(lds,         g);
  async_load_b128(lds +  512u, g +  512u);
  async_load_b128(lds + 1024u, g + 1024u);
}

__device__ __forceinline__ void async_store_row(uint64_t g, uint32_t lds) {
  async_store_b128(g,          lds);
  async_store_b128(g +  512u,  lds +  512u);
  async_store_b128(g + 1024u,  lds + 1024u);
}

// ---- Kernel 1: per-batch inclusive scan (wave32 shuffle scan, 2 barriers) ----

__global__ void __launch_bounds__(T_)
lr_cumsum(const int* __restrict__ dur, int* __restrict__ cum,
          float* __restrict__ mel_out) {
  const int b    = blockIdx.x;
  const int t    = threadIdx.x;
  const int wave = t >> 5;
  const int lane = t & 31;
  __shared__ int s_wsum[T_ / 32];    // 16 wave totals

  int v = dur[b * T_ + t];
#pragma unroll
  for (int off = 1; off < 32; off <<= 1) {   // intra-wave inclusive scan
    int n = __shfl_up(v, off, 32);
    if (lane >= off) v += n;
  }
  if (lane == 31) s_wsum[wave] = v;
  __syncthreads();
  if (wave == 0) {                           // scan the 16 wave totals
    int w = (lane < 16) ? s_wsum[lane] : 0;
#pragma unroll
    for (int off = 1; off < 16; off <<= 1) {
      int n = __shfl_up(w, off, 32);
      if (lane >= off) w += n;
    }
    if (lane < 16) s_wsum[lane] = w;         // inclusive prefix of wave sums
  }
  __syncthreads();
  if (wave > 0) v += s_wsum[wave - 1];
  cum[b * T_ + t] = v;
  if (t == T_ - 1) mel_out[b] = (float)v;    // mel_len output (tail of d_out)
}

// ---- Kernel 2: expansion. One wave per row; double-buffered async pipeline ----

__global__ void __launch_bounds__(256)
lr_expand(const float* __restrict__ x, const int* __restrict__ cum,
          float* __restrict__ out) {
  __shared__ __align__(16) int   s_cum[T_];             // 2 KB: search table
  __shared__ __align__(16) float s_zero[D_];            // 1.5 KB: zero row
  __shared__ __align__(16) float s_buf[WAVES][2][D_];   // 24 KB: double buffers

  const int b    = blockIdx.x / (ML_ / POS_PER_BLOCK);
  const int tile = blockIdx.x % (ML_ / POS_PER_BLOCK);
  const int t    = threadIdx.x;

  for (int i = t; i < T_; i += 256) s_cum[i]  = cum[b * T_ + i];
  for (int i = t; i < D_; i += 256) s_zero[i] = 0.0f;
  __syncthreads();

  const int wave = t >> 5;
  const int lane = t & 31;
  const int mel  = s_cum[T_ - 1];
  const uint32_t lane16 = (uint32_t)lane * 16u;

  const uint32_t zoff    = lds_off_u32(&s_zero[0]) + lane16;
  const uint32_t slot0   = lds_off_u32(&s_buf[wave][0][0]) + lane16;
  const uint32_t slot1   = lds_off_u32(&s_buf[wave][1][0]) + lane16;
  const uint64_t out_base = (uint64_t)(uintptr_t)(out + (size_t)b * ML_ * D_);
  const uint64_t x_base   = (uint64_t)(uintptr_t)(x   + (size_t)b * T_  * D_);

  // upper_bound(cum, p): smallest i with cum[i] > p (searchsorted 'right').
  // Wave-uniform, so EXEC stays all-ones at every async call site.
  auto find_src = [&](int p) -> uint64_t {
    int lo = 0, hi = T_;
    while (lo < hi) {
      int mid = (lo + hi) >> 1;
      if (s_cum[mid] <= p) lo = mid + 1; else hi = mid;
    }
    const int idx = (lo < T_) ? lo : (T_ - 1);
    return x_base + (uint64_t)idx * ROW_BYTES + lane16;
  };

  const int p0 = tile * POS_PER_BLOCK + wave * 8;   // wave-uniform start row

  // Prologue: prime slot 0 with row p0 if it is valid.
  if (p0 < mel) async_load_row(slot0, find_src(p0));

#pragma unroll 1
  for (int k = 0; k < 8; ++k) {
    const int  p          = p0 + k;
    const bool valid      = (p < mel);                 // prefix property:
    const bool next_valid = (k < 7) && (p + 1 < mel);  // valid rows first
    uint64_t next_src = 0;
    if (next_valid) next_src = find_src(p + 1);        // search overlaps waits

    // One wait per valid row: retires this row's loads AND previous row's
    // stores together (load/store latencies overlap across iterations).
    if (valid) wait_async0();

    const uint64_t dst   = out_base + (uint64_t)p * ROW_BYTES + lane16;
    const uint32_t src_s = valid ? ((k & 1) ? slot1 : slot0) : zoff;
    async_store_row(dst, src_s);

    if (next_valid) async_load_row((k & 1) ? slot0 : slot1, next_src);
    // Zero tail: no loads, no waits -- pure async streaming from zoff.
  }
  // s_endpgm performs an implicit wait-idle; all outstanding stores retire.
}

extern "C" void kernel_launch(void* const* d_in, const int* in_sizes, int n_in,
                              void* d_out, int out_size, void* d_ws, size_t ws_size,
                              hipStream_t stream) {
  const float* x   = (const float*)d_in[0];
  const int*   dur = (const int*)d_in[1];   // durations (values 0..7)
  // d_in[2] = max_len scalar; fixed at compile time (2048)

  float* out     = (float*)d_out;
  float* mel_out = out + (size_t)B_ * ML_ * D_;   // tuple output #2, flat tail
  int*   cum     = (int*)d_ws;                    // B*T ints = 64 KB scratch

  lr_cumsum<<<B_, T_, 0, stream>>>(dur, cum, mel_out);
  lr_expand<<<B_ * (ML_ / POS_PER_BLOCK), 256, 0, stream>>>(x, cum, out);
}